// CRFLoss_15659450761457
// MI455X (gfx1250) — compile-verified
//
#include <hip/hip_runtime.h>
#include <hip/hip_bf16.h>

typedef __attribute__((ext_vector_type(2))) float v2f;
typedef __attribute__((ext_vector_type(8))) float v8f;
typedef __attribute__((ext_vector_type(4))) unsigned int u32x4;
typedef __attribute__((ext_vector_type(8))) int i32x8;
typedef __attribute__((ext_vector_type(4))) int i32x4;

#define Bb 64
#define Ss 512
#define Kk 48
#define T2 510              // number of recurrence steps (t-1 with t=511)
#define CH 16               // chunk length
#define NCK 32              // ceil(510/16)
#define LSTRIDE 50          // LDS row stride (floats), avoids bank conflicts
#define TILE_ELTS (Kk*Kk)   // 2304

// workspace layout (in floats)
#define WS_P  0                         // B*NCK*K*K chunk product matrices
#define WS_C  (Bb*NCK*Kk*Kk)            // B*NCK chunk log-scales
#define WS_GP (WS_C + Bb*NCK)           // 64 gold partials
#define WS_FE (WS_GP + 64)              // 64 per-batch fs[end]

#if defined(__has_builtin)
#if __has_builtin(__builtin_amdgcn_tensor_load_to_lds) && \
    __has_builtin(__builtin_amdgcn_s_wait_tensorcnt)
#define USE_TDM 1
#endif
#endif
#ifndef USE_TDM
#define USE_TDM 0
#endif

#if USE_TDM
// Issue a TDM DMA of one flat 2304-float tile (9216 B) global -> LDS.
// D# group0: [1:0]=count=1, [63:32]=lds_addr, [120:64]=global_addr, [127:126]=type=2
// D# group1: data_size=4B, tensor_dim0=tile_dim0=2304, tensor_dim1=tile_dim1=1,
//            tensor_dim0_stride=2304
__device__ __forceinline__ void tdm_load_tile(const float* gsrc, unsigned lds_off)
{
    unsigned long long ga = (unsigned long long)(uintptr_t)gsrc;
    u32x4 g0;
    g0.x = 1u;                                                   // count=1
    g0.y = lds_off;                                              // lds_addr
    g0.z = (unsigned)ga;                                         // global_addr[31:0]
    g0.w = (unsigned)((ga >> 32) & 0x01FFFFFFu) | 0x80000000u;   // ga[56:32] | type=2
    i32x8 g1;
    g1[0] = 0x20000;                    // data_size = 2 (4 bytes)
    g1[1] = (int)(2304u << 16);         // tensor_dim0[15:0] in bits[63:48]
    g1[2] = 0x10000;                    // tensor_dim1 = 1 (bits[111:80])
    g1[3] = (int)(2304u << 16);         // tile_dim0 = 2304 (bits[127:112])
    g1[4] = 1;                          // tile_dim1 = 1
    g1[5] = 2304;                       // tensor_dim0_stride[31:0]
    g1[6] = 0;
    g1[7] = 0;
    i32x4 z4 = {0, 0, 0, 0};
    i32x8 z8 = {0, 0, 0, 0, 0, 0, 0, 0};
    // clang-23 / therock-10.0 headers: 6-arg form
    __builtin_amdgcn_tensor_load_to_lds(g0, g1, z4, z4, z8, 0);
}
#endif

// ---------------------------------------------------------------------------
// Phase 1: per (batch, chunk) compute P = exp(S_{i_last}) * ... * exp(S_{i0})
// with global-max rescaling each step; log-scale accumulated in csum.
// 288 threads = 9 wave32s; wave w owns 16x16 output tile (w/3, w%3).
// Matmul D = E * P via V_WMMA_F32_16X16X4_F32 (full f32 precision).
// Raw tiles double-buffered in LDS via TDM (tensor_load_to_lds), overlapping
// the next tile's DMA with the current step's WMMA chain.
// ---------------------------------------------------------------------------
__global__ __launch_bounds__(288) void crf_phase1(
    const float* __restrict__ scores, const int* __restrict__ lengths,
    float* __restrict__ Pout, float* __restrict__ Cout)
{
    __shared__ float Raw[2][TILE_ELTS];     // raw score tiles (DMA targets)
    __shared__ float E[Kk * LSTRIDE];       // exp(tile), padded
    __shared__ float Pm[Kk * LSTRIDE];      // running product, padded
    __shared__ float red[12];

    const int b    = blockIdx.y;
    const int c    = blockIdx.x;
    const int tid  = threadIdx.x;
    const int w    = tid >> 5;          // wave id 0..8
    const int lane = tid & 31;
    const int ti   = w / 3, tj = w % 3; // output tile coords
    const int lm   = lane & 15;         // lane % 16
    const int lh   = lane >> 4;         // lane / 16

    // init running product P = identity
    for (int e = tid; e < Kk * Kk; e += 288) {
        int r = e / Kk, col = e - r * Kk;
        Pm[r * LSTRIDE + col] = (r == col) ? 1.0f : 0.0f;
    }
    float csum = 0.0f;

    const int len1 = lengths[b] - 1;    // reference: lengths - 1
    const int i0   = c * CH;
    int iend = len1 - 1;                // step i active iff i < len1-1
    if (iend < i0) iend = i0;
    int ich_end = i0 + CH;
    if (ich_end > T2) ich_end = T2;
    if (iend > ich_end) iend = ich_end;

    const float* base_tile = scores + ((size_t)b * Ss + 2) * TILE_ELTS;

#if USE_TDM
    if (w == 0 && i0 < iend) {
        tdm_load_tile(base_tile + (size_t)i0 * TILE_ELTS,
                      (unsigned)(uintptr_t)&Raw[0][0]);
    }
#endif
    __syncthreads();

    int buf = 0;
    for (int i = i0; i < iend; ++i) {
#if USE_TDM
        // overlap: DMA next tile into the other buffer, then wait for current
        if (w == 0) {
            if (i + 1 < iend) {
                tdm_load_tile(base_tile + (size_t)(i + 1) * TILE_ELTS,
                              (unsigned)(uintptr_t)&Raw[buf ^ 1][0]);
                __builtin_amdgcn_s_wait_tensorcnt((short)1);
            } else {
                __builtin_amdgcn_s_wait_tensorcnt((short)0);
            }
        }
        __syncthreads();
        // exponentiate Raw[buf] into padded E (vector LDS reads: 2 x b128/thread)
        {
            const float4* rv = (const float4*)&Raw[buf][0];
            float4 x0 = rv[tid * 2 + 0];
            float4 x1 = rv[tid * 2 + 1];
            int e0 = tid * 8;
            float vals[8] = {x0.x, x0.y, x0.z, x0.w, x1.x, x1.y, x1.z, x1.w};
            #pragma unroll
            for (int q = 0; q < 8; ++q) {
                int e = e0 + q;
                int r = e / Kk, col = e - r * Kk;
                E[r * LSTRIDE + col] = __expf(vals[q]);
            }
        }
#else
        const float* tile = base_tile + (size_t)i * TILE_ELTS;
        #pragma unroll
        for (int q = 0; q < 8; ++q) {
            int e = tid + q * 288;
            int r = e / Kk, col = e - r * Kk;
            E[r * LSTRIDE + col] = __expf(tile[e]);
        }
        if (i + 1 < iend && tid < 72) {
            __builtin_prefetch(base_tile + (size_t)(i + 1) * TILE_ELTS
                               + (size_t)tid * 32, 0, 0);
        }
#endif
        __syncthreads();

        // D = E * P  (48x48x48) : 12 k-steps of V_WMMA_F32_16X16X4_F32
        v8f acc = {};
        #pragma unroll
        for (int kk = 0; kk < 12; ++kk) {
            const int ka = kk * 4 + 2 * lh;         // A/B K-slot for this half-wave
            v2f a, bf;
            a.x  = E[(ti * 16 + lm) * LSTRIDE + ka];        // A[M][K], 16x4 layout
            a.y  = E[(ti * 16 + lm) * LSTRIDE + ka + 1];
            bf.x = Pm[ka * LSTRIDE + tj * 16 + lm];         // B[K][N], 4x16 layout
            bf.y = Pm[(ka + 1) * LSTRIDE + tj * 16 + lm];
            acc = __builtin_amdgcn_wmma_f32_16x16x4_f32(
                false, a, false, bf, (short)0, acc, false, false);
        }

        // global max of D for rescaling (all entries positive)
        float mx = 0.0f;
        #pragma unroll
        for (int v = 0; v < 8; ++v) mx = fmaxf(mx, acc[v]);
        #pragma unroll
        for (int off = 16; off > 0; off >>= 1)
            mx = fmaxf(mx, __shfl_xor(mx, off, 32));
        if (lane == 0) red[w] = mx;
        __syncthreads();                // also fences all reads of Pm
        float s = red[0];
        #pragma unroll
        for (int q = 1; q < 9; ++q) s = fmaxf(s, red[q]);
        s = fmaxf(s, 1e-30f);
        const float inv = 1.0f / s;
        csum += __logf(s);

        // P <- D / s   (C/D 16x16 layout: VGPR v -> M = v + 8*lh)
        #pragma unroll
        for (int v = 0; v < 8; ++v)
            Pm[(ti * 16 + v + 8 * lh) * LSTRIDE + tj * 16 + lm] = acc[v] * inv;
        __syncthreads();
        buf ^= 1;
    }

    float* Po = Pout + ((size_t)b * NCK + c) * (Kk * Kk);
    for (int e = tid; e < Kk * Kk; e += 288) {
        int r = e / Kk, col = e - r * Kk;
        Po[e] = Pm[r * LSTRIDE + col];
    }
    if (tid == 0) Cout[b * NCK + c] = csum;
}

// ---------------------------------------------------------------------------
// Gold score: sum of scores[b, t+1, targets[b,t+1]] for t < lengths[b]-1
// ---------------------------------------------------------------------------
__global__ __launch_bounds__(256) void crf_gold(
    const float* __restrict__ scores, const int* __restrict__ targets,
    const int* __restrict__ lengths, float* __restrict__ gp)
{
    __shared__ float sred[256];
    float acc = 0.0f;
    const int total = Bb * (Ss - 1);
    for (int idx = blockIdx.x * 256 + threadIdx.x; idx < total; idx += 64 * 256) {
        int b = idx / (Ss - 1);
        int t = idx - b * (Ss - 1);
        int len1 = lengths[b] - 1;
        if (t < len1) {
            int tg = targets[b * Ss + t + 1];
            acc += scores[((size_t)b * Ss + (size_t)(t + 1)) * (Kk * Kk) + tg];
        }
    }
    sred[threadIdx.x] = acc;
    __syncthreads();
    for (int off = 128; off > 0; off >>= 1) {
        if (threadIdx.x < off) sred[threadIdx.x] += sred[threadIdx.x + off];
        __syncthreads();
    }
    if (threadIdx.x == 0) gp[blockIdx.x] = sred[0];
}

// ---------------------------------------------------------------------------
// Phase 2: per batch, fold the 32 chunk maps into fs0 via mat-vec; tiny.
// ---------------------------------------------------------------------------
__global__ __launch_bounds__(64) void crf_phase2(
    const float* __restrict__ scores, const int* __restrict__ lengths,
    const float* __restrict__ Pin, const float* __restrict__ Cin,
    const int* __restrict__ start_ind, const int* __restrict__ end_ind,
    float* __restrict__ fe)
{
    __shared__ float p[Kk];
    __shared__ float wv[Kk];
    const int b = blockIdx.x;
    const int j = threadIdx.x;
    const int si = start_ind[0];
    const int ei = end_ind[0];

    // fs0 = scores[b, 1, :, start_ind]
    if (j < Kk) p[j] = scores[((size_t)b * Ss + 1) * (Kk * Kk) + j * Kk + si];
    __syncthreads();
    float mx = -1e30f;
    for (int m = 0; m < Kk; ++m) mx = fmaxf(mx, p[m]);
    float base = mx;
    __syncthreads();
    if (j < Kk) p[j] = __expf(p[j] - mx);
    __syncthreads();

    for (int cc = 0; cc < NCK; ++cc) {
        const float* Pc = Pin + ((size_t)b * NCK + cc) * (Kk * Kk);
        if (j < Kk) {
            float acc = 0.0f;
            for (int m = 0; m < Kk; ++m) acc += Pc[j * Kk + m] * p[m];
            wv[j] = acc;
        }
        __syncthreads();
        float s = 0.0f;
        for (int m = 0; m < Kk; ++m) s = fmaxf(s, wv[m]);
        s = fmaxf(s, 1e-30f);
        base += Cin[b * NCK + cc] + __logf(s);
        __syncthreads();
        if (j < Kk) p[j] = wv[j] / s;
        __syncthreads();
    }
    if (j == 0) fe[b] = base + __logf(fmaxf(p[ei], 1e-38f));
}

// ---------------------------------------------------------------------------
// Finalize: out = sum_b fs_end[b] - sum gold partials (deterministic)
// ---------------------------------------------------------------------------
__global__ void crf_finalize(const float* __restrict__ gp,
                             const float* __restrict__ fe,
                             float* __restrict__ out)
{
    if (threadIdx.x == 0) {
        float g = 0.0f, a = 0.0f;
        for (int i = 0; i < 64; ++i) { g += gp[i]; a += fe[i]; }
        out[0] = a - g;
    }
}

extern "C" void kernel_launch(void* const* d_in, const int* in_sizes, int n_in,
                              void* d_out, int out_size, void* d_ws, size_t ws_size,
                              hipStream_t stream) {
    const float* scores    = (const float*)d_in[0];
    const int*   targets   = (const int*)d_in[1];
    const int*   lengths   = (const int*)d_in[2];
    const int*   start_ind = (const int*)d_in[3];
    const int*   end_ind   = (const int*)d_in[4];

    float* ws   = (float*)d_ws;
    float* Pbuf = ws + WS_P;
    float* Cbuf = ws + WS_C;
    float* gp   = ws + WS_GP;
    float* fe   = ws + WS_FE;

    crf_phase1<<<dim3(NCK, Bb), 288, 0, stream>>>(scores, lengths, Pbuf, Cbuf);
    crf_gold<<<64, 256, 0, stream>>>(scores, targets, lengths, gp);
    crf_phase2<<<Bb, 64, 0, stream>>>(scores, lengths, Pbuf, Cbuf,
                                      start_ind, end_ind, fe);
    crf_finalize<<<1, 64, 0, stream>>>(gp, fe, (float*)d_out);
}